// FFTBias2d_55602646614163
// MI455X (gfx1250) — compile-verified
//
#include <hip/hip_runtime.h>

typedef __attribute__((ext_vector_type(2))) float v2f;
typedef __attribute__((ext_vector_type(8))) float v8f;

// Shapes: B=8, S=4096, H=16, E=64, s=64.  bh = b*16+h (128 slabs).
// ws layout (floats): vs[128*4096] us[...] Rv[...] Ru[...] zrow[16*64]

// ---------------- Kernel A: row/col sums over the 64x64 grid ----------------
// One block per (b,h). Thread: e4-group = tx&15 (4 consecutive e), g = tx>>4
// handles j in {g, g+16, g+32, g+48}. float4 global loads -> 256B contiguous
// per 16-lane group. Column sums register-blocked; row sums reduced via LDS.
__global__ void __launch_bounds__(256) reduce_rc_kernel(const float* __restrict__ v,
                                                        float* __restrict__ vs,
                                                        float* __restrict__ us) {
  const int bh = blockIdx.x;
  const int b = bh >> 4, h = bh & 15;
  const int tx = threadIdx.x;
  const int c16 = tx & 15;   // which float4 of the 64 e's
  const int g   = tx >> 4;   // j subgroup 0..15
  const float4* v4 = (const float4*)v;
  float4* vs4 = (float4*)vs;
  float4* us4 = (float4*)us;

  __shared__ float4 rowpart[256];   // [g][c16]

  float4 col[4];
#pragma unroll
  for (int m = 0; m < 4; ++m) { col[m].x = col[m].y = col[m].z = col[m].w = 0.f; }

  for (int i = 0; i < 64; ++i) {
    float4 r; r.x = r.y = r.z = r.w = 0.f;
#pragma unroll
    for (int m = 0; m < 4; ++m) {
      const int j = g + 16 * m;
      const int ss = i * 64 + j;
      float4 val = v4[((b * 4096 + ss) * 16 + h) * 16 + c16];
      r.x += val.x; r.y += val.y; r.z += val.z; r.w += val.w;
      col[m].x += val.x; col[m].y += val.y; col[m].z += val.z; col[m].w += val.w;
    }
    rowpart[g * 16 + c16] = r;
    __syncthreads();
    if (tx < 16) {                       // 16 threads own the 64 e's (as float4)
      float4 u; u.x = u.y = u.z = u.w = 0.f;
#pragma unroll
      for (int gg = 0; gg < 16; ++gg) {
        float4 p = rowpart[gg * 16 + tx];
        u.x += p.x; u.y += p.y; u.z += p.z; u.w += p.w;
      }
      us4[(bh * 64 + i) * 16 + tx] = u;  // u_s[b,h][i][e]
    }
    __syncthreads();
  }
#pragma unroll
  for (int m = 0; m < 4; ++m) {
    const int j = g + 16 * m;
    vs4[(bh * 64 + j) * 16 + c16] = col[m]; // v_s[b,h][j][e]
  }
}

// ------------- Kernel B: Toeplitz 64x64x64 GEMMs via V_WMMA_F32_16X16X4_F32 -------------
// One wave32 block per (b,h).  R[n][e] = sum_j T[n][j] * M[j][e],
// T[n][j] = w[h][(n-j+64) mod 127].  A=T tile (16x4, M x K), B=M tile (4x16, K x N).
__global__ void __launch_bounds__(32) toeplitz_wmma_kernel(const float* __restrict__ w,
                                                           const float* __restrict__ o_,
                                                           const float* __restrict__ vs,
                                                           const float* __restrict__ us,
                                                           float* __restrict__ Rv,
                                                           float* __restrict__ Ru,
                                                           float* __restrict__ zrow) {
  const int bh = blockIdx.x;
  const int b = bh >> 4, h = bh & 15;
  const int lane = threadIdx.x;
  const int l16 = lane & 15;
  const int koff = (lane >> 4) << 1;          // lanes 0-15 -> K 0,1 ; lanes 16-31 -> K 2,3

  __shared__ float Tl[64 * 64];
  __shared__ float Ml[2][64 * 64];

  for (int idx = lane; idx < 4096; idx += 32) {
    const int n = idx >> 6, j = idx & 63;
    int m = n - j + 64; if (m >= 127) m -= 127; // Toeplitz index into w[h][0..126]
    Tl[idx] = w[h * 127 + m];
    Ml[0][idx] = vs[bh * 4096 + idx];
    Ml[1][idx] = us[bh * 4096 + idx];
  }
  __syncthreads();

  float* const dsts[2] = { Rv, Ru };
#pragma unroll
  for (int mat = 0; mat < 2; ++mat) {
    const float* M = Ml[mat];
    float* dst = dsts[mat];
    for (int tn = 0; tn < 4; ++tn) {
      for (int te = 0; te < 4; ++te) {
        v8f c = {};
#pragma unroll
        for (int k4 = 0; k4 < 16; ++k4) {
          const int j0 = k4 * 4 + koff;
          v2f a, bb;
          a.x  = Tl[(tn * 16 + l16) * 64 + j0];       // A: row n, K j0
          a.y  = Tl[(tn * 16 + l16) * 64 + j0 + 1];   //    row n, K j0+1
          bb.x = M[j0 * 64 + te * 16 + l16];          // B: K j0,  col e
          bb.y = M[(j0 + 1) * 64 + te * 16 + l16];    //    K j0+1, col e
          c = __builtin_amdgcn_wmma_f32_16x16x4_f32(
              false, a, false, bb, (short)0, c, false, false);
        }
        const int ebase = te * 16 + l16;
        const int nbase = tn * 16 + ((lane >> 4) << 3); // C/D: VGPR r -> M=r (+8 upper lanes)
#pragma unroll
        for (int r = 0; r < 8; ++r)
          dst[bh * 4096 + (nbase + r) * 64 + ebase] = c[r];
      }
    }
  }

  if (b == 0) {                 // z_pb row vector: zrow[h][n] = s * sum_j o_[j]*T[n][j]
    for (int n = lane; n < 64; n += 32) {
      float acc = 0.f;
      for (int j = 0; j < 64; ++j) acc += o_[j] * Tl[n * 64 + j];
      zrow[h * 64 + n] = 64.0f * acc;
    }
  }
}

// ---------------- Kernel C: broadcast-add writer (134 MB stream-out) ----------------
// One block per (b,i): out[b, i*64+j, h, e] = Rv[b,h][j][e] + Ru[b,h][i][e].
__global__ void __launch_bounds__(256) writer_kernel(const float* __restrict__ Rv,
                                                     const float* __restrict__ Ru,
                                                     float* __restrict__ out) {
  const int blk = blockIdx.x;        // b*64 + i
  const int b = blk >> 6, i = blk & 63;
  const int tx = threadIdx.x;
  const float4* Rv4 = (const float4*)Rv;
  const float4* Ru4 = (const float4*)Ru;
  float4* out4 = (float4*)out;
  __shared__ float4 RuL[256];        // [h][e4] for this fixed i
  {
    const int h = tx >> 4, c = tx & 15;
    RuL[tx] = Ru4[((b * 16 + h) * 64 + i) * 16 + c];
  }
  __syncthreads();
  for (int t = tx; t < 16384; t += 256) {   // 64(j)*16(h)*16(float4 of e)
    const int c = t & 15;
    const int h = (t >> 4) & 15;
    const int j = t >> 8;
    float4 a = Rv4[((b * 16 + h) * 64 + j) * 16 + c];
    float4 u = RuL[h * 16 + c];
    float4 o; o.x = a.x + u.x; o.y = a.y + u.y; o.z = a.z + u.z; o.w = a.w + u.w;
    out4[((b * 4096 + i * 64 + j) * 16 + h) * 16 + c] = o;
  }
}

// ---------------- Kernel D: z_pb output (1, S, H) ----------------
__global__ void __launch_bounds__(256) zpb_kernel(const float* __restrict__ zrow,
                                                  float* __restrict__ out2) {
  const int flat = blockIdx.x * 256 + threadIdx.x;  // == ss*16 + h
  if (flat < 65536) {
    const int h = flat & 15;
    const int ss = flat >> 4;
    const int j = ss & 63, i = ss >> 6;
    out2[flat] = zrow[h * 64 + j] + zrow[h * 64 + i];
  }
}

extern "C" void kernel_launch(void* const* d_in, const int* in_sizes, int n_in,
                              void* d_out, int out_size, void* d_ws, size_t ws_size,
                              hipStream_t stream) {
  (void)in_sizes; (void)n_in; (void)out_size; (void)ws_size;
  const float* v  = (const float*)d_in[0];   // (8,4096,16,64)
  const float* w  = (const float*)d_in[1];   // (1,16,127)
  const float* o_ = (const float*)d_in[2];   // (64,)
  float* out = (float*)d_out;

  float* vs   = (float*)d_ws;                // 128*4096 each
  float* us   = vs + 524288;
  float* Rv   = us + 524288;
  float* Ru   = Rv + 524288;
  float* zrow = Ru + 524288;                 // 16*64

  reduce_rc_kernel   <<<128, 256, 0, stream>>>(v, vs, us);
  toeplitz_wmma_kernel<<<128,  32, 0, stream>>>(w, o_, vs, us, Rv, Ru, zrow);
  writer_kernel      <<<512, 256, 0, stream>>>(Rv, Ru, out);
  zpb_kernel         <<<256, 256, 0, stream>>>(zrow, out + 33554432);
}